// peaks_InfoNCE_47665547051144
// MI455X (gfx1250) — compile-verified
//
#include <hip/hip_runtime.h>
#include <math.h>

#define B_    256
#define C_    768
#define H_    24
#define W_    24
#define HW_   576
#define EDIM  2304   // 3*C

typedef __attribute__((ext_vector_type(2))) float v2f;
typedef __attribute__((ext_vector_type(8))) float v8f;

// ---------------------------------------------------------------------------
// Kernel A: channel mean -> LDS heatmap -> 3x3 clipped max/min pooling ->
// peak flags (bit0 = max peak, bit1 = min peak) + inverse peak counts.
// Grid: 512 blocks (tensor*256 + batch), 576 threads (one per pixel, 18 waves)
// ---------------------------------------------------------------------------
__global__ void __launch_bounds__(576)
kern_hm_masks(const float* __restrict__ f1, const float* __restrict__ f2,
              unsigned int* __restrict__ flags, float* __restrict__ invcnt) {
    __shared__ float hm[HW_];
    __shared__ int cmax, cmin;
    const int bb = blockIdx.x;          // [0,512)
    const int t  = bb >> 8;
    const int b  = bb & 255;
    const float* feat = (t == 0 ? f1 : f2) + (size_t)b * C_ * HW_;
    const int s = threadIdx.x;          // pixel index [0,576)

    float sum = 0.0f;
    for (int c = 0; c < C_; ++c)        // coalesced across s for each c
        sum += feat[(size_t)c * HW_ + s];
    hm[s] = sum * (1.0f / (float)C_);
    if (threadIdx.x == 0) { cmax = 0; cmin = 0; }
    __syncthreads();

    const int h = s / W_, w = s % W_;
    const float v = hm[s];
    float mx = v, mn = v;
    const int h0 = (h > 0) ? h - 1 : 0, h1 = (h < H_ - 1) ? h + 1 : h;
    const int w0 = (w > 0) ? w - 1 : 0, w1 = (w < W_ - 1) ? w + 1 : w;
    for (int hh = h0; hh <= h1; ++hh)
        for (int ww = w0; ww <= w1; ++ww) {
            const float u = hm[hh * W_ + ww];
            mx = fmaxf(mx, u);
            mn = fminf(mn, u);
        }
    const bool corner = (h == 0 || h == H_ - 1) && (w == 0 || w == W_ - 1);
    unsigned int fl = 0;
    if (!corner && v >= mx) { fl |= 1u; atomicAdd(&cmax, 1); }
    if (!corner && v <= mn) { fl |= 2u; atomicAdd(&cmin, 1); }
    flags[(size_t)bb * HW_ + s] = fl;
    __syncthreads();
    if (threadIdx.x == 0) {
        invcnt[bb * 2 + 0] = 1.0f / (float)cmax;
        invcnt[bb * 2 + 1] = 1.0f / (float)cmin;
    }
}

// ---------------------------------------------------------------------------
// Kernel B: masked pooled sums -> raw embedding rows E[bb][0:768]=f_max,
// [768:1536]=f_min, [1536:2304]=f_gap.  Grid: 512 blocks x 512 thr (16 waves,
// one wave per channel, coalesced spatial reads, wave-shuffle reduction).
// ---------------------------------------------------------------------------
__global__ void __launch_bounds__(512)
kern_pool(const float* __restrict__ f1, const float* __restrict__ f2,
          const unsigned int* __restrict__ flags,
          const float* __restrict__ invcnt, float* __restrict__ E) {
    __shared__ unsigned int sfl[HW_];
    const int bb = blockIdx.x;
    const int t  = bb >> 8;
    const int b  = bb & 255;
    const float* feat = (t == 0 ? f1 : f2) + (size_t)b * C_ * HW_;
    for (int i = threadIdx.x; i < HW_; i += blockDim.x)
        sfl[i] = flags[(size_t)bb * HW_ + i];
    __syncthreads();

    const float imax = invcnt[bb * 2 + 0];
    const float imin = invcnt[bb * 2 + 1];
    const int wave = threadIdx.x >> 5;
    const int lane = threadIdx.x & 31;
    float* Erow = E + (size_t)bb * EDIM;

    for (int c = wave; c < C_; c += 16) {
        const float* src = feat + (size_t)c * HW_;
        float smax = 0.0f, smin = 0.0f, sgap = 0.0f;
        for (int j = lane; j < HW_; j += 32) {   // 18 coalesced loads per lane
            const float v = src[j];
            const unsigned int fl = sfl[j];
            sgap += v;
            if (fl & 1u) smax += v;
            if (fl & 2u) smin += v;
        }
        for (int off = 16; off >= 1; off >>= 1) {
            smax += __shfl_xor(smax, off, 32);
            smin += __shfl_xor(smin, off, 32);
            sgap += __shfl_xor(sgap, off, 32);
        }
        if (lane == 0) {
            Erow[c]          = smax * imax;
            Erow[C_ + c]     = smin * imin;
            Erow[2 * C_ + c] = sgap * (1.0f / (float)HW_);
        }
    }
}

// ---------------------------------------------------------------------------
// Kernel C: L2-normalize each 2304-element embedding row in place.
// Grid: 512 blocks x 256 threads (each thread owns 9 elements).
// ---------------------------------------------------------------------------
__global__ void __launch_bounds__(256)
kern_norm(float* __restrict__ E) {
    __shared__ float red[8];
    float* row = E + (size_t)blockIdx.x * EDIM;
    float vals[9];
    float ss = 0.0f;
    #pragma unroll
    for (int i = 0; i < 9; ++i) {
        vals[i] = row[threadIdx.x + i * 256];
        ss += vals[i] * vals[i];
    }
    for (int off = 16; off >= 1; off >>= 1) ss += __shfl_xor(ss, off, 32);
    const int wave = threadIdx.x >> 5, lane = threadIdx.x & 31;
    if (lane == 0) red[wave] = ss;
    __syncthreads();
    if (threadIdx.x == 0) {
        float tot = 0.0f;
        for (int i = 0; i < 8; ++i) tot += red[i];
        red[0] = rsqrtf(tot);
    }
    __syncthreads();
    const float inv = red[0];
    #pragma unroll
    for (int i = 0; i < 9; ++i) row[threadIdx.x + i * 256] = vals[i] * inv;
}

// ---------------------------------------------------------------------------
// Kernel D: logits = scale * E1 @ E2^T via V_WMMA_F32_16X16X4_F32.
// Grid: (16,16) blocks x 32 threads; one wave computes a 16x16 f32 tile,
// K = 2304 -> 576 WMMA issues. A fragment (16x4): lanes 0-15 hold K=k,k+1 of
// row M=lane; lanes 16-31 hold K=k+2,k+3 of row M=lane-16. B fragment (4x16)
// is symmetric with N across lanes, so both sides use identical float2 loads.
// ---------------------------------------------------------------------------
__global__ void __launch_bounds__(32)
kern_gemm(const float* __restrict__ E, const float* __restrict__ scale_p,
          float* __restrict__ logits) {
    const float* E1 = E;
    const float* E2 = E + (size_t)B_ * EDIM;
    const int tm = blockIdx.x, tn = blockIdx.y;
    const int lane = threadIdx.x;
    const int hl = lane & 15;       // M (for A) / N (for B)
    const int hi = lane >> 4;       // selects K pair (0,1) vs (2,3)
    const float* arow = E1 + (size_t)(tm * 16 + hl) * EDIM + 2 * hi;
    const float* brow = E2 + (size_t)(tn * 16 + hl) * EDIM + 2 * hi;

    v8f acc = {0.f, 0.f, 0.f, 0.f, 0.f, 0.f, 0.f, 0.f};
    for (int k = 0; k < EDIM; k += 4) {
        const v2f a = *(const v2f*)(arow + k);
        const v2f b = *(const v2f*)(brow + k);
        acc = __builtin_amdgcn_wmma_f32_16x16x4_f32(
            /*neg_a=*/false, a, /*neg_b=*/false, b,
            /*c_mod=*/(short)0, acc, /*reuse_a=*/false, /*reuse_b=*/false);
    }
    const float sc = scale_p[0];
    #pragma unroll
    for (int r = 0; r < 8; ++r) {
        const int row = tm * 16 + r + 8 * hi;   // D layout: VGPR r -> M=r(+8)
        const int col = tn * 16 + hl;           // lane -> N
        logits[row * B_ + col] = acc[r] * sc;
    }
}

// ---------------------------------------------------------------------------
// Kernel E: row/column log-sum-exp of logits + diagonal extraction.
// Grid: 512 blocks x 256 threads (block i<256: row i; else column i-256).
// ---------------------------------------------------------------------------
__global__ void __launch_bounds__(256)
kern_lse(const float* __restrict__ logits, float* __restrict__ lse,
         float* __restrict__ diag) {
    __shared__ float red[8];
    const int bi = blockIdx.x;
    const bool isRow = bi < B_;
    const int idx = isRow ? bi : bi - B_;
    const int tid = threadIdx.x;
    const float x = isRow ? logits[idx * B_ + tid] : logits[tid * B_ + idx];
    if (isRow && tid == idx) diag[idx] = x;

    float m = x;
    for (int off = 16; off >= 1; off >>= 1) m = fmaxf(m, __shfl_xor(m, off, 32));
    const int wave = tid >> 5, lane = tid & 31;
    if (lane == 0) red[wave] = m;
    __syncthreads();
    float bm = red[0];
    #pragma unroll
    for (int i = 1; i < 8; ++i) bm = fmaxf(bm, red[i]);
    __syncthreads();

    float e = __expf(x - bm);
    for (int off = 16; off >= 1; off >>= 1) e += __shfl_xor(e, off, 32);
    if (lane == 0) red[wave] = e;
    __syncthreads();
    if (tid == 0) {
        float s = 0.0f;
        for (int i = 0; i < 8; ++i) s += red[i];
        lse[bi] = bm + logf(s);
    }
}

// ---------------------------------------------------------------------------
// Kernel F: loss = ( mean_i(lse_row_i - d_i) + mean_i(lse_col_i - d_i) ) / 2
// ---------------------------------------------------------------------------
__global__ void __launch_bounds__(256)
kern_loss(const float* __restrict__ lse, const float* __restrict__ diag,
          float* __restrict__ out) {
    __shared__ float red[8];
    const int tid = threadIdx.x;
    float v = (lse[tid] - diag[tid]) + (lse[B_ + tid] - diag[tid]);
    for (int off = 16; off >= 1; off >>= 1) v += __shfl_xor(v, off, 32);
    const int wave = tid >> 5, lane = tid & 31;
    if (lane == 0) red[wave] = v;
    __syncthreads();
    if (tid == 0) {
        float s = 0.0f;
        for (int i = 0; i < 8; ++i) s += red[i];
        out[0] = s * (1.0f / (2.0f * (float)B_));
    }
}

// ---------------------------------------------------------------------------
extern "C" void kernel_launch(void* const* d_in, const int* in_sizes, int n_in,
                              void* d_out, int out_size, void* d_ws, size_t ws_size,
                              hipStream_t stream) {
    const float* f1    = (const float*)d_in[0];
    const float* f2    = (const float*)d_in[1];
    const float* scale = (const float*)d_in[2];
    float* out = (float*)d_out;

    // Workspace carve-up (all 256-byte aligned)
    char* ws = (char*)d_ws;
    size_t off = 0;
    auto take = [&](size_t bytes) -> char* {
        char* p = ws + off;
        off += (bytes + 255) & ~(size_t)255;
        return p;
    };
    unsigned int* flags  = (unsigned int*)take((size_t)2 * B_ * HW_ * 4); // 1.18 MB
    float*        invcnt = (float*)take((size_t)2 * B_ * 2 * 4);
    float*        E      = (float*)take((size_t)2 * B_ * EDIM * 4);      // 4.72 MB
    float*        logits = (float*)take((size_t)B_ * B_ * 4);            // 0.26 MB
    float*        lse    = (float*)take((size_t)2 * B_ * 4);
    float*        diag   = (float*)take((size_t)B_ * 4);
    (void)ws_size; (void)in_sizes; (void)n_in; (void)out_size;

    kern_hm_masks<<<dim3(2 * B_), dim3(576), 0, stream>>>(f1, f2, flags, invcnt);
    kern_pool    <<<dim3(2 * B_), dim3(512), 0, stream>>>(f1, f2, flags, invcnt, E);
    kern_norm    <<<dim3(2 * B_), dim3(256), 0, stream>>>(E);
    kern_gemm    <<<dim3(16, 16), dim3(32),  0, stream>>>(E, scale, logits);
    kern_lse     <<<dim3(2 * B_), dim3(256), 0, stream>>>(logits, lse, diag);
    kern_loss    <<<dim3(1),      dim3(256), 0, stream>>>(lse, diag, out);
}